// GammaCapsule_13211319402523
// MI455X (gfx1250) — compile-verified
//
#include <hip/hip_runtime.h>
#include <math.h>

// Problem constants (reference: B,K,J,D,E = 64,32,1152,16,8; 3 routings)
#define Bn 64
#define Kn 32
#define Jn 1152
#define Dn 16
#define En 8
#define WAVES 8          // waves per block in the WMMA kernels
#define NPARTS (128 * WAVES)
#define PFJ 64           // W prefetch distance in j (8 own-stride iterations)

typedef __attribute__((ext_vector_type(2))) float v2f;
typedef __attribute__((ext_vector_type(8))) float v8f;

// -------- u_norm[b,j] = ||inputs[b,j,:]||, stored transposed (j-major) ------
__global__ void unorm_kernel(const float* __restrict__ inp, float* __restrict__ unT) {
  int idx = blockIdx.x * blockDim.x + threadIdx.x;   // 0..B*J-1 (exact grid)
  int b = idx / Jn, j = idx - b * Jn;
  const float4* p = (const float4*)(inp + ((size_t)b * Jn + j) * En);
  float4 x = p[0], y = p[1];
  float q = x.x*x.x + x.y*x.y + x.z*x.z + x.w*x.w
          + y.x*y.x + y.y*y.y + y.z*y.z + y.w*y.w;
  unT[(size_t)j * Bn + b] = __builtin_amdgcn_sqrtf(q);
}

// -------- s_j kernel: one block per (b-block of 16, capsule k) --------------
// Transposed tile: C[d, b] = W(16d x 8e) x inputs^T(8e x 16b), two chained
// V_WMMA_F32_16X16X4_F32. 8 waves stride j; fixed-order LDS tree reduction.
template<bool UNIFORM_C>
__global__ void s_kernel(const float* __restrict__ inp, const float* __restrict__ W,
                         const float* __restrict__ unT, const float* __restrict__ ct,
                         float* __restrict__ s) {
  const int tid  = threadIdx.x;
  const int lane = tid & 31;
  const int wid  = tid >> 5;
  const int k    = blockIdx.x & 31;
  const int bblk = (blockIdx.x >> 5) * 16;
  const int lm   = lane & 15;          // column b offset (and A-row d)
  const int half = lane >> 4;
  const int e0   = 2 * half;

  __shared__ float red[8 * 256];       // [g*256 + tid], conflict-free

  const v8f zero = {0,0,0,0,0,0,0,0};
  v8f acc = zero;

  const float* Wbase = W + (((size_t)k * Jn) * Dn + lm) * En;   // + j*Dn*En
  const float* Ibase = inp + ((size_t)(bblk + lm) * Jn) * En;   // + j*En

  for (int j = wid; j < Jn; j += WAVES) {
    const float* wrow = Wbase + (size_t)j * (Dn * En);
    const float* arow = Ibase + (size_t)j * En;
    int jp = (j + PFJ < Jn) ? (j + PFJ) : j;
    __builtin_prefetch(Wbase + (size_t)jp * (Dn * En), 0, 3);

    v2f a0 = *(const v2f*)(wrow + e0);        // A = W tile, row d=lm
    v2f a1 = *(const v2f*)(wrow + e0 + 4);
    v2f b0 = *(const v2f*)(arow + e0);        // B = inputs^T, col b=lm
    v2f b1 = *(const v2f*)(arow + e0 + 4);
    v8f t = __builtin_amdgcn_wmma_f32_16x16x4_f32(false, a0, false, b0,
                                                  (short)0, zero, false, false);
    t     = __builtin_amdgcn_wmma_f32_16x16x4_f32(false, a1, false, b1,
                                                  (short)0, t,    false, false);

    // column (b) sum of squares over d: lane-local over 8 VGPRs, + xor16
    float q = t[0] * t[0];
#pragma unroll
    for (int g = 1; g < 8; ++g) q = fmaf(t[g], t[g], q);
    q += __shfl_xor(q, 16, 32);

    float un = unT[(size_t)j * Bn + bblk + lm];                    // coalesced
    float w  = UNIFORM_C ? (1.0f / 32.0f)
                         : ct[((size_t)j * Kn + k) * Bn + bblk + lm];
    // min(nrm,un)/nrm * c == min(1, un*rsqrt(q)) * c
    float f = fminf(1.0f, un * __builtin_amdgcn_rsqf(q)) * w;
#pragma unroll
    for (int g = 0; g < 8; ++g) acc[g] = fmaf(t[g], f, acc[g]);
  }

  // ---- fixed-order cross-wave reduction in LDS ----
#pragma unroll
  for (int g = 0; g < 8; ++g) red[g * 256 + tid] = acc[g];
  __syncthreads();
  if (tid < 128) {
#pragma unroll
    for (int g = 0; g < 8; ++g) red[g * 256 + tid] += red[g * 256 + tid + 128];
  }
  __syncthreads();
  if (tid < 64) {
#pragma unroll
    for (int g = 0; g < 8; ++g) red[g * 256 + tid] += red[g * 256 + tid + 64];
  }
  __syncthreads();
  if (tid < 32) {
#pragma unroll
    for (int g = 0; g < 8; ++g) {
      float r = red[g * 256 + tid] + red[g * 256 + tid + 32];
      int d = g + 8 * half;                                  // lane = tid here
      s[((size_t)(bblk + lm) * Kn + k) * Dn + d] = r;
    }
  }
}

// -------- squash: v = n2/(1+n2) * s/sqrt(n2+1e-9), s += bias first ----------
__global__ void squash_kernel(const float* __restrict__ s, const float* __restrict__ bias,
                              float* __restrict__ v) {
  int idx = blockIdx.x * blockDim.x + threadIdx.x;  // 0..B*K-1 (exact)
  int k = idx & 31;
  float tmp[16];
  float n2 = 0.f;
#pragma unroll
  for (int d = 0; d < 16; ++d) {
    float x = s[(size_t)idx * Dn + d] + bias[k * Dn + d];
    tmp[d] = x;
    n2 += x * x;
  }
  float sc = (n2 / (1.f + n2)) * __builtin_amdgcn_rsqf(n2 + 1e-9f);
#pragma unroll
  for (int d = 0; d < 16; ++d) v[(size_t)idx * Dn + d] = tmp[d] * sc;
}

// -------- distance kernel: d[b,k,j] = ||v[b,k,:] - u_hat_clip[b,k,j,:]|| ----
__global__ void dist_kernel(const float* __restrict__ inp, const float* __restrict__ W,
                            const float* __restrict__ unT, const float* __restrict__ v,
                            float* __restrict__ dmat, float* __restrict__ parts) {
  const int tid  = threadIdx.x;
  const int lane = tid & 31;
  const int wid  = tid >> 5;
  const int k    = blockIdx.x & 31;
  const int bblk = (blockIdx.x >> 5) * 16;
  const int lm   = lane & 15;
  const int half = lane >> 4;
  const int e0   = 2 * half;

  const v8f zero = {0,0,0,0,0,0,0,0};

  float vload[8];                        // v[b=bblk+lm, k, d=g+8*half]
#pragma unroll
  for (int g = 0; g < 8; ++g)
    vload[g] = v[((size_t)(bblk + lm) * Kn + k) * Dn + g + 8 * half];

  const float* Wbase = W + (((size_t)k * Jn) * Dn + lm) * En;
  const float* Ibase = inp + ((size_t)(bblk + lm) * Jn) * En;

  float lsum = 0.f;                      // all 32 lanes: each column counted 2x
  for (int j = wid; j < Jn; j += WAVES) {
    const float* wrow = Wbase + (size_t)j * (Dn * En);
    const float* arow = Ibase + (size_t)j * En;
    int jp = (j + PFJ < Jn) ? (j + PFJ) : j;
    __builtin_prefetch(Wbase + (size_t)jp * (Dn * En), 0, 3);

    v2f a0 = *(const v2f*)(wrow + e0);
    v2f a1 = *(const v2f*)(wrow + e0 + 4);
    v2f b0 = *(const v2f*)(arow + e0);
    v2f b1 = *(const v2f*)(arow + e0 + 4);
    v8f t = __builtin_amdgcn_wmma_f32_16x16x4_f32(false, a0, false, b0,
                                                  (short)0, zero, false, false);
    t     = __builtin_amdgcn_wmma_f32_16x16x4_f32(false, a1, false, b1,
                                                  (short)0, t,    false, false);

    float q = t[0] * t[0];
#pragma unroll
    for (int g = 1; g < 8; ++g) q = fmaf(t[g], t[g], q);
    q += __shfl_xor(q, 16, 32);

    float un = unT[(size_t)j * Bn + bblk + lm];
    float f  = fminf(1.0f, un * __builtin_amdgcn_rsqf(q));

    float e00 = t[0] * f - vload[0];
    float qq = e00 * e00;
#pragma unroll
    for (int g = 1; g < 8; ++g) {
      float e = t[g] * f - vload[g];
      qq = fmaf(e, e, qq);
    }
    qq += __shfl_xor(qq, 16, 32);        // full D=16 distance^2, both halves
    float dr = __builtin_amdgcn_sqrtf(qq);
    lsum += dr;                          // branch-free; halved at the end
    if (half == 0)                       // single writer per column b
      dmat[((size_t)(bblk + lm) * Kn + k) * Jn + j] = dr;
  }
  // fixed-order full-wave butterfly; every column appeared twice -> * 0.5
  lsum += __shfl_xor(lsum, 1, 32);
  lsum += __shfl_xor(lsum, 2, 32);
  lsum += __shfl_xor(lsum, 4, 32);
  lsum += __shfl_xor(lsum, 8, 32);
  lsum += __shfl_xor(lsum, 16, 32);
  if (lane == 0) parts[blockIdx.x * WAVES + wid] = lsum * 0.5f;
}

// -------- t = t_const / (EPS - 0.5 * mean(d)); fixed-order deterministic sum -
__global__ void t_kernel(const float* __restrict__ parts, float* __restrict__ tptr) {
  if (threadIdx.x == 0 && blockIdx.x == 0) {
    float sm = 0.f;
    for (int i = 0; i < NPARTS; ++i) sm += parts[i];
    float mean = sm / (float)(Bn * Kn * Jn);
    const float t_const = 5.6312117818213f;   // log(0.9*31) - log(0.1) = log(279)
    tptr[0] = t_const / (1e-12f - 0.5f * mean);
  }
}

// -------- softmax over k of b_ij = t*d; writes output-layout c and j-major ct
__global__ void softmax_kernel(const float* __restrict__ dmat, const float* __restrict__ tptr,
                               float* __restrict__ c_out, float* __restrict__ ct) {
  const int lane = threadIdx.x & 31;                       // lane = k
  const int wid  = blockIdx.x * 8 + (threadIdx.x >> 5);    // 0..B*J-1 (exact)
  int b = wid / Jn, j = wid - b * Jn;
  float t = tptr[0];
  float x = t * dmat[((size_t)b * Kn + lane) * Jn + j];
  float m = x;
  m = fmaxf(m, __shfl_xor(m, 1, 32));
  m = fmaxf(m, __shfl_xor(m, 2, 32));
  m = fmaxf(m, __shfl_xor(m, 4, 32));
  m = fmaxf(m, __shfl_xor(m, 8, 32));
  m = fmaxf(m, __shfl_xor(m, 16, 32));
  float e = __expf(x - m);
  float sm = e;
  sm += __shfl_xor(sm, 1, 32);
  sm += __shfl_xor(sm, 2, 32);
  sm += __shfl_xor(sm, 4, 32);
  sm += __shfl_xor(sm, 8, 32);
  sm += __shfl_xor(sm, 16, 32);
  float c = e * __builtin_amdgcn_rcpf(sm);
  c_out[((size_t)b * Kn + lane) * Jn + j] = c;
  ct[((size_t)j * Kn + lane) * Bn + b] = c;
}

extern "C" void kernel_launch(void* const* d_in, const int* in_sizes, int n_in,
                              void* d_out, int out_size, void* d_ws, size_t ws_size,
                              hipStream_t stream) {
  (void)in_sizes; (void)n_in; (void)out_size; (void)ws_size;
  const float* inp  = (const float*)d_in[0];   // (B,J,E)
  const float* W    = (const float*)d_in[1];   // (1,K,J,D,E)
  const float* bias = (const float*)d_in[2];   // (1,K,D)
  float* out = (float*)d_out;                  // v_j (B,K,D) ++ c_ij (B,K,J,1)
  float* ws  = (float*)d_ws;

  float* unT   = ws;                 // 73728   : u_norm, j-major
  float* dmat  = unT + 73728;        // 2359296 : d[b,k,j]
  float* sbuf  = dmat + 2359296;     // 32768   : s_j
  float* vbuf  = sbuf + 32768;       // 32768   : v_j (intermediate)
  float* ct    = vbuf + 32768;       // 2359296 : c, j-major for coalesced reads
  float* parts = ct + 2359296;       // 1024    : per-wave partial d sums
  float* tptr  = parts + NPARTS;     // 1       : routing temperature t

  float* v_out = out;                // 32768 floats
  float* c_out = out + 32768;        // 2359296 floats

  unorm_kernel<<<288, 256, 0, stream>>>(inp, unT);

  // r = 0 (c uniform = 1/32)
  s_kernel<true><<<128, 256, 0, stream>>>(inp, W, unT, nullptr, sbuf);
  squash_kernel<<<8, 256, 0, stream>>>(sbuf, bias, vbuf);
  dist_kernel<<<128, 256, 0, stream>>>(inp, W, unT, vbuf, dmat, parts);
  t_kernel<<<1, 32, 0, stream>>>(parts, tptr);

  // r = 1
  softmax_kernel<<<9216, 256, 0, stream>>>(dmat, tptr, c_out, ct);
  s_kernel<false><<<128, 256, 0, stream>>>(inp, W, unT, ct, sbuf);
  squash_kernel<<<8, 256, 0, stream>>>(sbuf, bias, vbuf);
  dist_kernel<<<128, 256, 0, stream>>>(inp, W, unT, vbuf, dmat, parts);
  t_kernel<<<1, 32, 0, stream>>>(parts, tptr);

  // r = 2 (final c_ij and v_j)
  softmax_kernel<<<9216, 256, 0, stream>>>(dmat, tptr, c_out, ct);
  s_kernel<false><<<128, 256, 0, stream>>>(inp, W, unT, ct, sbuf);
  squash_kernel<<<8, 256, 0, stream>>>(sbuf, bias, v_out);
}